// GraphEncoder_54065048322841
// MI455X (gfx1250) — compile-verified
//
#include <hip/hip_runtime.h>

typedef __attribute__((ext_vector_type(2))) float v2f;
typedef __attribute__((ext_vector_type(8))) float v8f;

#define COUT 64

// ---------------- utility kernels ----------------

__global__ void k_fill(float* __restrict__ p, float v, long long n) {
    long long g = (long long)blockIdx.x * blockDim.x + threadIdx.x;
    if (g < n) p[g] = v;
}

// deg[dst] += ew  (deg pre-initialized to 1.0 == self loop weight)
__global__ void k_deg(const long long* __restrict__ ei, const float* __restrict__ ew,
                      float* __restrict__ deg, long long E) {
    long long e = (long long)blockIdx.x * blockDim.x + threadIdx.x;
    if (e < E) atomicAdd(&deg[ei[E + e]], ew[e]);
}

// deg -> d^{-1/2} in place
__global__ void k_rsqrt(float* __restrict__ d, long long n) {
    long long g = (long long)blockIdx.x * blockDim.x + threadIdx.x;
    if (g < n) { float v = d[g]; d[g] = (v > 0.f) ? rsqrtf(v) : 0.f; }
}

// norm[e] = dis[src] * ew[e] * dis[dst]
__global__ void k_norm(const long long* __restrict__ ei, const float* __restrict__ ew,
                       const float* __restrict__ dis, float* __restrict__ nrm, long long E) {
    long long e = (long long)blockIdx.x * blockDim.x + threadIdx.x;
    if (e < E) nrm[e] = dis[ei[e]] * ew[e] * dis[ei[E + e]];
}

// ---------------- WMMA GEMM: T = X @ W  (X: n x CIN, W: CIN x 64) ----------------
// Block = 256 threads = 8 waves. Each wave computes a 16-row x 64-col tile
// via 4 WMMA_F32_16X16X4_F32 accumulators over K = CIN.

template <int CIN>
__global__ void k_gemm(const float* __restrict__ X, const float* __restrict__ W,
                       float* __restrict__ T, int n) {
    constexpr int XS = CIN + 4;                 // LDS row pad: avoid 16-way bank conflict
    __shared__ float sW[CIN * COUT];            // CIN x 64
    __shared__ float sX[8][16 * XS];            // per-wave 16 x CIN tile

    const int tid  = threadIdx.x;
    const int wave = tid >> 5;
    const int lane = tid & 31;
    const int row0 = blockIdx.x * 128 + wave * 16;

    for (int i = tid; i < CIN * COUT; i += 256) sW[i] = W[i];

    float* xs = &sX[wave][0];
    for (int i = lane; i < 16 * CIN; i += 32) {
        int r = i / CIN, c = i - r * CIN;
        int gr = row0 + r;
        xs[r * XS + c] = (gr < n) ? X[(long long)gr * CIN + c] : 0.f;
    }
    __syncthreads();

    v8f acc0 = {}, acc1 = {}, acc2 = {}, acc3 = {};
    const int m     = lane & 15;        // A row / B col-in-tile / D col
    const int khalf = (lane >> 4) * 2;  // lanes 0-15: K+0,K+1 ; lanes 16-31: K+2,K+3

#pragma unroll
    for (int k = 0; k < CIN; k += 4) {
        v2f a;
        a.x = xs[m * XS + k + khalf];
        a.y = xs[m * XS + k + khalf + 1];
        const float* w0 = &sW[(k + khalf) * COUT];
        const float* w1 = &sW[(k + khalf + 1) * COUT];
        v2f b0, b1, b2, b3;
        b0.x = w0[m +  0]; b0.y = w1[m +  0];
        b1.x = w0[m + 16]; b1.y = w1[m + 16];
        b2.x = w0[m + 32]; b2.y = w1[m + 32];
        b3.x = w0[m + 48]; b3.y = w1[m + 48];
        acc0 = __builtin_amdgcn_wmma_f32_16x16x4_f32(false, a, false, b0, (short)0, acc0, false, false);
        acc1 = __builtin_amdgcn_wmma_f32_16x16x4_f32(false, a, false, b1, (short)0, acc1, false, false);
        acc2 = __builtin_amdgcn_wmma_f32_16x16x4_f32(false, a, false, b2, (short)0, acc2, false, false);
        acc3 = __builtin_amdgcn_wmma_f32_16x16x4_f32(false, a, false, b3, (short)0, acc3, false, false);
    }

    // D layout: VGPR j holds row M = j + 8*(lane>=16), col N = lane&15 (per 16-tile)
    const int mbase = (lane >> 4) * 8;
#pragma unroll
    for (int j = 0; j < 8; ++j) {
        int gr = row0 + mbase + j;
        if (gr < n) {
            float* out = &T[(long long)gr * COUT];
            out[m +  0] = acc0[j];
            out[m + 16] = acc1[j];
            out[m + 32] = acc2[j];
            out[m + 48] = acc3[j];
        }
    }
}

// ---------------- message scatter: agg[dst] += norm * t[src] ----------------
// one thread per (edge, channel): 32 lanes cover 32 contiguous channels -> coalesced
__global__ void k_scatter(const float* __restrict__ t, const long long* __restrict__ ei,
                          const float* __restrict__ nrm, float* __restrict__ agg, long long E) {
    long long g = (long long)blockIdx.x * blockDim.x + threadIdx.x;
    if (g >= E * COUT) return;
    long long e = g >> 6;
    int c = (int)(g & 63);
    long long s = ei[e], d = ei[E + e];
    atomicAdd(&agg[d * COUT + c], nrm[e] * t[s * COUT + c]);
}

// h = silu(agg + dis^2 * t + b)   (self-loop folded in, no atomics needed)
__global__ void k_finalize(float* __restrict__ h, const float* __restrict__ t,
                           const float* __restrict__ dis, const float* __restrict__ b,
                           long long n) {
    long long g = (long long)blockIdx.x * blockDim.x + threadIdx.x;
    if (g >= n * COUT) return;
    long long ni = g >> 6;
    int c = (int)(g & 63);
    float dd = dis[ni];
    float v = h[g] + dd * dd * t[g] + b[c];
    h[g] = v * (1.f / (1.f + __expf(-v)));
}

// ---------------- pooling ----------------

__global__ void k_poolsum(const float* __restrict__ h, const long long* __restrict__ batch,
                          float* __restrict__ out, long long n) {
    long long g = (long long)blockIdx.x * blockDim.x + threadIdx.x;
    if (g >= n * COUT) return;
    long long ni = g >> 6;
    int c = (int)(g & 63);
    atomicAdd(&out[batch[ni] * COUT + c], h[g]);
}

__global__ void k_poolcnt(const long long* __restrict__ batch, float* __restrict__ cnt, long long n) {
    long long g = (long long)blockIdx.x * blockDim.x + threadIdx.x;
    if (g < n) atomicAdd(&cnt[batch[g]], 1.f);
}

__global__ void k_pooldiv(float* __restrict__ out, const float* __restrict__ cnt, int G) {
    int g = blockIdx.x * blockDim.x + threadIdx.x;
    if (g < G * COUT) out[g] /= fmaxf(cnt[g >> 6], 1.f);
}

// ---------------- driver ----------------

static inline unsigned gridFor(long long n, int bs) { return (unsigned)((n + bs - 1) / bs); }

extern "C" void kernel_launch(void* const* d_in, const int* in_sizes, int n_in,
                              void* d_out, int out_size, void* d_ws, size_t ws_size,
                              hipStream_t stream) {
    const float*     x     = (const float*)d_in[0];
    const float*     ew    = (const float*)d_in[1];
    const float*     W1    = (const float*)d_in[2];
    const float*     b1    = (const float*)d_in[3];
    const float*     W2    = (const float*)d_in[4];
    const float*     b2    = (const float*)d_in[5];
    const float*     W3    = (const float*)d_in[6];
    const float*     b3    = (const float*)d_in[7];
    const long long* ei    = (const long long*)d_in[8];
    const long long* batch = (const long long*)d_in[9];
    float* out = (float*)d_out;

    const long long N = in_sizes[0] / 32;
    const long long E = in_sizes[1];
    const int       G = out_size / COUT;

    float* ws     = (float*)d_ws;
    float* dis    = ws;                               // N
    float* nrm    = dis + N;                          // E
    float* buf0   = nrm + E;                          // N*64
    float* buf1   = buf0 + (size_t)N * COUT;          // N*64
    float* counts = buf1 + (size_t)N * COUT;          // G

    const int BS = 256;

    // --- GCN normalization (shared across layers) ---
    k_fill <<<gridFor(N, BS), BS, 0, stream>>>(dis, 1.f, N);          // self-loop deg
    k_deg  <<<gridFor(E, BS), BS, 0, stream>>>(ei, ew, dis, E);
    k_rsqrt<<<gridFor(N, BS), BS, 0, stream>>>(dis, N);
    k_norm <<<gridFor(E, BS), BS, 0, stream>>>(ei, ew, dis, nrm, E);

    const unsigned gGemm = gridFor(N, 128);
    const unsigned gNC   = gridFor(N * COUT, BS);
    const unsigned gEC   = gridFor(E * COUT, BS);

    // --- layer 1: x(Nx32) -> buf1 ---
    k_gemm<32><<<gGemm, BS, 0, stream>>>(x, W1, buf0, (int)N);
    k_fill    <<<gNC, BS, 0, stream>>>(buf1, 0.f, N * COUT);
    k_scatter <<<gEC, BS, 0, stream>>>(buf0, ei, nrm, buf1, E);
    k_finalize<<<gNC, BS, 0, stream>>>(buf1, buf0, dis, b1, N);

    // --- layer 2: buf1 -> buf1 ---
    k_gemm<64><<<gGemm, BS, 0, stream>>>(buf1, W2, buf0, (int)N);
    k_fill    <<<gNC, BS, 0, stream>>>(buf1, 0.f, N * COUT);
    k_scatter <<<gEC, BS, 0, stream>>>(buf0, ei, nrm, buf1, E);
    k_finalize<<<gNC, BS, 0, stream>>>(buf1, buf0, dis, b2, N);

    // --- layer 3: buf1 -> buf1 ---
    k_gemm<64><<<gGemm, BS, 0, stream>>>(buf1, W3, buf0, (int)N);
    k_fill    <<<gNC, BS, 0, stream>>>(buf1, 0.f, N * COUT);
    k_scatter <<<gEC, BS, 0, stream>>>(buf0, ei, nrm, buf1, E);
    k_finalize<<<gNC, BS, 0, stream>>>(buf1, buf0, dis, b3, N);

    // --- global mean pool ---
    k_fill   <<<gridFor((long long)G * COUT, BS), BS, 0, stream>>>(out, 0.f, (long long)G * COUT);
    k_fill   <<<gridFor(G, BS), BS, 0, stream>>>(counts, 0.f, G);
    k_poolsum<<<gNC, BS, 0, stream>>>(buf1, batch, out, N);
    k_poolcnt<<<gridFor(N, BS), BS, 0, stream>>>(batch, counts, N);
    k_pooldiv<<<gridFor((long long)G * COUT, BS), BS, 0, stream>>>(out, counts, G);
}